// ModuleSoftsplat_22763326669391
// MI455X (gfx1250) — compile-verified
//
#include <hip/hip_runtime.h>

typedef float v4f __attribute__((ext_vector_type(4)));

// Problem constants (fixed by the reference): B=4, C=64, H=512, W=512
constexpr int B   = 4;
constexpr int C   = 64;
constexpr int H   = 512;
constexpr int W   = 512;
constexpr int HW  = H * W;        // 262144 = 2^18
constexpr int CHW = C * HW;       // 16777216

// Tiling for LDS privatization of the scatter-add
constexpr int TILE     = 32;                  // pixel tile side
constexpr int HALO     = 4;                   // covers |displacement| <~ 4 (flow ~ N(0,1))
constexpr int LT       = TILE + 2 * HALO;     // 40 (multiple of 4 -> b128 LDS ops)
constexpr int LTSZ     = LT * LT;             // 1600 floats per channel tile
constexpr int CPB      = 4;                   // channels per LDS pass (25.6 KB LDS)
constexpr int NTHREADS = 256;
constexpr int PPT      = 4;                   // 4 consecutive-x pixels per thread
constexpr int TPI      = (H / TILE) * (W / TILE);   // 256 tiles per image

// ---------------------------------------------------------------------------
// Kernel 1: zero the output (harness poisons d_out; scatter-add needs 0 base)
// ---------------------------------------------------------------------------
__global__ __launch_bounds__(256) void softsplat_zero(float4* __restrict__ out, int n4) {
    int i = blockIdx.x * 256 + threadIdx.x;
    if (i < n4) {
        float4 z; z.x = 0.f; z.y = 0.f; z.z = 0.f; z.w = 0.f;
        out[i] = z;  // coalesced global_store_b128
    }
}

// ---------------------------------------------------------------------------
// Kernel 2: forward summation splatting with LDS tile privatization.
//  - thread handles 4 consecutive-x pixels: image/flow reads are b128
//  - per pixel: flow/weights/indices computed once, reused for 64 channels
//  - in-halo corners -> ds_add_f32 into LDS tile
//  - out-of-halo corners (rare) -> global_atomic_add_f32 fallback
//  - per (tile, 4 channels): b128 zero, b128 flush with zero-skip + bounds
// ---------------------------------------------------------------------------
__global__ __launch_bounds__(NTHREADS) void softsplat_fwd(const float* __restrict__ image,
                                                          const float* __restrict__ flow,
                                                          float* __restrict__ out) {
    __shared__ __align__(16) float lds[CPB * LTSZ];

    const int bid = blockIdx.x;            // 0 .. B*TPI-1
    const int b   = bid >> 8;              // / TPI (256 tiles per image)
    const int t   = bid & (TPI - 1);
    const int ty0 = (t >> 4) * TILE;       // tile origin y
    const int tx0 = (t & 15) * TILE;       // tile origin x

    const int tid  = threadIdx.x;
    const int row  = tid >> 3;             // 0..31 : row within tile
    const int xblk = (tid & 7) * 4;        // 0,4,...,28 : 4-px block within row

    const int xp0 = tx0 + xblk;            // first pixel x (multiple of 4)
    const int yp  = ty0 + row;             // pixel y
    const int p0  = yp * W + xp0;          // first pixel linear index (16B aligned)

    // ---------------- per-pixel precompute (reused for all channels) --------
    const v4f fx4 = *(const v4f*)(flow + b * 2 * HW + p0);        // flow x plane
    const v4f fy4 = *(const v4f*)(flow + b * 2 * HW + HW + p0);   // flow y plane

    int      info[PPT];                    // intile: lds base idx ; else: y0*W + x0
    float    wt[PPT][4];                   // bilinear corner weights, k = dy*2 + dx
    unsigned flags = 0;                    // bit(8j)=intile, bits(8j+4+k)=valid

#pragma unroll
    for (int j = 0; j < PPT; ++j) {
        const float fx = fx4[j] + (float)(xp0 + j);
        const float fy = fy4[j] + (float)yp;

        const float x0f = floorf(fx);
        const float y0f = floorf(fy);
        const int   x0  = (int)x0f;
        const int   y0  = (int)y0f;
        const float wx1 = fx - x0f, wx0 = 1.0f - wx1;
        const float wy1 = fy - y0f, wy0 = 1.0f - wy1;

        wt[j][0] = wx0 * wy0;   // dy=0,dx=0
        wt[j][1] = wx1 * wy0;   // dy=0,dx=1
        wt[j][2] = wx0 * wy1;   // dy=1,dx=0
        wt[j][3] = wx1 * wy1;   // dy=1,dx=1

        const int lx0 = x0 - tx0 + HALO;
        const int ly0 = y0 - ty0 + HALO;
        const bool intile = ((unsigned)lx0 <= (unsigned)(LT - 2)) &
                            ((unsigned)ly0 <= (unsigned)(LT - 2));
        if (intile) {
            info[j] = ly0 * LT + lx0;
            flags  |= 1u << (8 * j);
        } else {
            info[j] = y0 * W + x0;
#pragma unroll
            for (int k = 0; k < 4; ++k) {
                const int xi = x0 + (k & 1);
                const int yi = y0 + (k >> 1);
                const bool valid = (xi >= 0) & (xi < W) & (yi >= 0) & (yi < H);
                if (valid) flags |= 1u << (8 * j + 4 + k);
            }
        }
    }

    const float* __restrict__ ib = image + b * CHW;   // + c*HW + pixel
    float* __restrict__       ob = out   + b * CHW;

    // ---------------- channel passes ---------------------------------------
    for (int cc = 0; cc < C; cc += CPB) {
        __syncthreads();   // previous flush done before re-zeroing LDS
        {
            const v4f zv = {0.0f, 0.0f, 0.0f, 0.0f};
            v4f* l4 = (v4f*)lds;
            for (int i = tid; i < (CPB * LTSZ) / 4; i += NTHREADS) l4[i] = zv;
        }
        __syncthreads();

        // accumulate CPB channels into LDS (fallback: global atomics)
#pragma unroll
        for (int u = 0; u < CPB; ++u) {
            const int c = cc + u;
            float* __restrict__ ltile = &lds[u * LTSZ];
            const float* __restrict__ ic = ib + c * HW;
            float* __restrict__       oc = ob + c * HW;

            // hide the 1 MB channel stride: prefetch next pass's channel
            if (cc + CPB < C)
                __builtin_prefetch(ib + (c + CPB) * HW + p0, 0, 3);

            // one b128 non-temporal load covers this thread's 4 pixels
            const v4f v4 = __builtin_nontemporal_load((const v4f*)(ic + p0));

#pragma unroll
            for (int j = 0; j < PPT; ++j) {
                const float v = v4[j];
                if (flags & (1u << (8 * j))) {
                    float* cell = ltile + info[j];
                    atomicAdd(cell,          v * wt[j][0]);  // ds_add_f32
                    atomicAdd(cell + 1,      v * wt[j][1]);
                    atomicAdd(cell + LT,     v * wt[j][2]);
                    atomicAdd(cell + LT + 1, v * wt[j][3]);
                } else {
#pragma unroll
                    for (int k = 0; k < 4; ++k) {
                        if (flags & (1u << (8 * j + 4 + k))) {
                            const int g = info[j] + (k >> 1) * W + (k & 1);
                            (void)__hip_atomic_fetch_add(oc + g, v * wt[j][k],
                                                         __ATOMIC_RELAXED,
                                                         __HIP_MEMORY_SCOPE_AGENT);
                        }
                    }
                }
            }
        }
        __syncthreads();

        // flush LDS tiles to global with atomics (b128 LDS reads, zero-skip)
#pragma unroll
        for (int u = 0; u < CPB; ++u) {
            const int c = cc + u;
            float* __restrict__ oc = ob + c * HW;
            const v4f* lt4 = (const v4f*)(&lds[u * LTSZ]);
            for (int i4 = tid; i4 < LTSZ / 4; i4 += NTHREADS) {
                const v4f val = lt4[i4];                  // ds_load_b128
                const int r   = i4 / (LT / 4);            // row in tile (div by 10)
                const int c4  = (i4 % (LT / 4)) * 4;      // col of first elem
                const int gy  = ty0 - HALO + r;
                const int gx0 = tx0 - HALO + c4;
                if ((unsigned)gy < (unsigned)H) {
                    float* orow = oc + gy * W;
#pragma unroll
                    for (int jj = 0; jj < 4; ++jj) {
                        if (val[jj] != 0.0f) {
                            const int gx = gx0 + jj;
                            if ((unsigned)gx < (unsigned)W) {
                                (void)__hip_atomic_fetch_add(orow + gx, val[jj],
                                                             __ATOMIC_RELAXED,
                                                             __HIP_MEMORY_SCOPE_AGENT);
                            }
                        }
                    }
                }
            }
        }
    }
}

// ---------------------------------------------------------------------------
extern "C" void kernel_launch(void* const* d_in, const int* in_sizes, int n_in,
                              void* d_out, int out_size, void* d_ws, size_t ws_size,
                              hipStream_t stream) {
    const float* image = (const float*)d_in[0];  // [B, C, H, W] f32
    const float* flow  = (const float*)d_in[1];  // [B, 2, H, W] f32
    float*       out   = (float*)d_out;          // [B, C, H, W] f32

    // Zero output: out_size = B*C*H*W floats -> float4 stores
    const int n4 = out_size / 4;
    softsplat_zero<<<(n4 + 255) / 256, 256, 0, stream>>>((float4*)out, n4);

    // Splat: one block per 32x32 pixel tile (B * 256 tiles = 1024 blocks)
    softsplat_fwd<<<B * TPI, NTHREADS, 0, stream>>>(image, flow, out);
}